// MLP_soft_CoSaMP_23270132810503
// MI455X (gfx1250) — compile-verified
//
#include <hip/hip_runtime.h>
#include <math.h>

#define BATCH 64
#define MDIM  512
#define NDIM  2048
#define SSP   16
#define KTOT  48                 // 16 rows Q_prev + 32 rows P_tilde (layer1: Q_prev = zero rows)
#define TAU_INV 10.0f
#define RIDGE 1e-6f

typedef __attribute__((ext_vector_type(2))) float v2f;
typedef __attribute__((ext_vector_type(8))) float v8f;

__device__ __forceinline__ v8f wmma4(v2f a, v2f b, v8f c) {
  // D(16x16,f32) += A(16x4,f32) * B(4x16,f32)
  return __builtin_amdgcn_wmma_f32_16x16x4_f32(false, a, false, b, (short)0, c,
                                               false, false);
}

// ---------------------------------------------------------------- init: X=0, P=0
__global__ void k_init(float* __restrict__ X, float* __restrict__ P) {
  size_t tid = (size_t)blockIdx.x * blockDim.x + threadIdx.x;
  size_t stride = (size_t)gridDim.x * blockDim.x;
  const size_t nP = (size_t)BATCH * KTOT * NDIM;
  for (size_t j = tid; j < nP; j += stride) P[j] = 0.0f;
  const size_t nX = (size_t)BATCH * NDIM;
  for (size_t j = tid; j < nX; j += stride) X[j] = 0.0f;
}

// ---------------------------------------------------------------- R = Y - X @ A^T
// D(64,512) = X(64,2048) @ A^T ; B[k][j] = A[j][k] (rows of A -> consecutive loads)
// One wave computes a 16x64 strip: 4 accumulators reuse each A-frag 4x.
__global__ void k_residual(const float* __restrict__ X, const float* __restrict__ A,
                           const float* __restrict__ Y, float* __restrict__ R) {
  const int wave = threadIdx.x >> 5, lane = threadIdx.x & 31;
  const int sid = blockIdx.x * 4 + wave;           // 0..31 strips
  const int tm = sid >> 3, tns = sid & 7;          // 4 M-tiles x 8 N-strips(64)
  const int half = lane >> 4, tl = lane & 15;
  const float* xr = X + (size_t)(tm * 16 + tl) * NDIM + 2 * half;
  const float* ar[4];
#pragma unroll
  for (int u = 0; u < 4; ++u)
    ar[u] = A + (size_t)(tns * 64 + u * 16 + tl) * NDIM + 2 * half;
  v8f acc[4] = {{}, {}, {}, {}};
  for (int k = 0; k < NDIM; k += 4) {
    const v2f a = *(const v2f*)(xr + k);
#pragma unroll
    for (int u = 0; u < 4; ++u)
      acc[u] = wmma4(a, *(const v2f*)(ar[u] + k), acc[u]);
  }
#pragma unroll
  for (int u = 0; u < 4; ++u) {
#pragma unroll
    for (int i = 0; i < 8; ++i) {
      int row = tm * 16 + i + 8 * half;
      int col = tns * 64 + u * 16 + tl;
      R[row * MDIM + col] = Y[row * MDIM + col] - acc[u][i];
    }
  }
}

// ---------------------------------------------------------------- SC = |W * (R @ A)|
// D(64,2048) = R(64,512) @ A(512,2048) ; B[k][n] = A[k][n] (strided column loads)
__global__ void k_scores(const float* __restrict__ R, const float* __restrict__ A,
                         const float* __restrict__ W, float* __restrict__ SC) {
  const int wave = threadIdx.x >> 5, lane = threadIdx.x & 31;
  const int sid = blockIdx.x * 4 + wave;           // 0..127 strips
  const int tm = sid >> 5, tns = sid & 31;         // 4 M-tiles x 32 N-strips(64)
  const int half = lane >> 4, tl = lane & 15;
  const float* rr = R + (size_t)(tm * 16 + tl) * MDIM + 2 * half;
  int col[4];
#pragma unroll
  for (int u = 0; u < 4; ++u) col[u] = tns * 64 + u * 16 + tl;
  v8f acc[4] = {{}, {}, {}, {}};
  for (int k = 0; k < MDIM; k += 4) {
    const v2f a = *(const v2f*)(rr + k);
    const float* ak = A + (size_t)(k + 2 * half) * NDIM;
#pragma unroll
    for (int u = 0; u < 4; ++u) {
      v2f b;
      b.x = ak[col[u]];
      b.y = ak[NDIM + col[u]];
      acc[u] = wmma4(a, b, acc[u]);
    }
  }
#pragma unroll
  for (int u = 0; u < 4; ++u) {
#pragma unroll
    for (int i = 0; i < 8; ++i) {
      int row = tm * 16 + i + 8 * half;
      SC[(size_t)row * NDIM + col[u]] = fabsf(W[col[u]] * acc[u][i]);
    }
  }
}

// ---------------------------------------------------------------- C_j = sum_k |s_j - s_k|
// Stages the per-sample score vector into LDS with GLOBAL_LOAD_ASYNC_TO_LDS_B128.
__global__ void k_pairwise(const float* __restrict__ SC, float* __restrict__ CC) {
  __shared__ float s[NDIM];
  const int b = blockIdx.x;
  const float* sb = SC + (size_t)b * NDIM;
  {
    // two async 16-byte copies per thread: 256 * 2 * 16B = 8KB tile
    unsigned dst0 = (unsigned)(size_t)(&s[threadIdx.x * 4]);
    unsigned long long g0 = (unsigned long long)(size_t)(sb + threadIdx.x * 4);
    asm volatile("global_load_async_to_lds_b128 %0, %1, off"
                 :: "v"(dst0), "v"(g0) : "memory");
    unsigned dst1 = (unsigned)(size_t)(&s[1024 + threadIdx.x * 4]);
    unsigned long long g1 = (unsigned long long)(size_t)(sb + 1024 + threadIdx.x * 4);
    asm volatile("global_load_async_to_lds_b128 %0, %1, off"
                 :: "v"(dst1), "v"(g1) : "memory");
    asm volatile("s_wait_asynccnt 0x0" ::: "memory");
  }
  __syncthreads();
  for (int j = threadIdx.x; j < NDIM; j += blockDim.x) {
    float sj = s[j], acc = 0.0f;
    for (int kk = 0; kk < NDIM; ++kk) acc += fabsf(sj - s[kk]);
    CC[(size_t)b * NDIM + j] = acc;
  }
}

// ---------------------------------------------------------------- softmax rows of NeuralSort
// row i (logical): logits_j = (coef_i * s_j - C_j)/tau ; writes P[b][rowOffset+i][:]
__global__ void k_softsort(const float* __restrict__ SC, const float* __restrict__ CC,
                           float* __restrict__ P, int rows, int rowOffset) {
  const int b = blockIdx.x / rows;
  const int i = blockIdx.x % rows;
  const float coef = 2047.0f - 2.0f * (float)i;    // n+1-2(i+1), n=2048
  const float* sb = SC + (size_t)b * NDIM;
  const float* cb = CC + (size_t)b * NDIM;
  __shared__ float red[256];
  // pass 1: max
  float mx = -3.0e38f;
  for (int j = threadIdx.x; j < NDIM; j += 256) {
    float l = (coef * sb[j] - cb[j]) * TAU_INV;
    mx = fmaxf(mx, l);
  }
  red[threadIdx.x] = mx;
  __syncthreads();
  for (int st = 128; st > 0; st >>= 1) {
    if (threadIdx.x < st) red[threadIdx.x] = fmaxf(red[threadIdx.x], red[threadIdx.x + st]);
    __syncthreads();
  }
  mx = red[0];
  __syncthreads();
  // pass 2: sum of exp
  float sm = 0.0f;
  for (int j = threadIdx.x; j < NDIM; j += 256) {
    float l = (coef * sb[j] - cb[j]) * TAU_INV;
    sm += __expf(l - mx);
  }
  red[threadIdx.x] = sm;
  __syncthreads();
  for (int st = 128; st > 0; st >>= 1) {
    if (threadIdx.x < st) red[threadIdx.x] += red[threadIdx.x + st];
    __syncthreads();
  }
  float inv = 1.0f / red[0];
  // pass 3: write normalized row
  float* dst = P + ((size_t)b * KTOT + rowOffset + i) * NDIM;
  for (int j = threadIdx.x; j < NDIM; j += 256) {
    float l = (coef * sb[j] - cb[j]) * TAU_INV;
    dst[j] = __expf(l - mx) * inv;
  }
}

// ---------------------------------------------------------------- MM_b = P_b @ A^T  (48x512 per sample)
// One wave computes a 16x64 strip (4 accumulators).
__global__ void k_pmat(const float* __restrict__ P, const float* __restrict__ A,
                       float* __restrict__ MM) {
  const int wave = threadIdx.x >> 5, lane = threadIdx.x & 31;
  const int gw = blockIdx.x * 4 + wave;            // 0..1535
  const int b = gw / 24, t = gw % 24;
  const int tm = t >> 3, tns = t & 7;              // 3 M-tiles x 8 N-strips(64)
  const int half = lane >> 4, tl = lane & 15;
  const float* pr = P + ((size_t)b * KTOT + tm * 16 + tl) * NDIM + 2 * half;
  const float* ar[4];
#pragma unroll
  for (int u = 0; u < 4; ++u)
    ar[u] = A + (size_t)(tns * 64 + u * 16 + tl) * NDIM + 2 * half;
  v8f acc[4] = {{}, {}, {}, {}};
  for (int k = 0; k < NDIM; k += 4) {
    __builtin_prefetch(ar[0] + k + 256, 0, 1);     // global_prefetch_b8 into L2
    const v2f a = *(const v2f*)(pr + k);
#pragma unroll
    for (int u = 0; u < 4; ++u)
      acc[u] = wmma4(a, *(const v2f*)(ar[u] + k), acc[u]);
  }
#pragma unroll
  for (int u = 0; u < 4; ++u) {
#pragma unroll
    for (int i = 0; i < 8; ++i) {
      int row = tm * 16 + i + 8 * half;
      int col = tns * 64 + u * 16 + tl;
      MM[((size_t)b * KTOT + row) * MDIM + col] = acc[u][i];
    }
  }
}

// ---------------------------------------------------------------- G_b = MM_b @ MM_b^T + ridge*I
__global__ void k_gram(const float* __restrict__ MM, float* __restrict__ G) {
  const int wave = threadIdx.x >> 5, lane = threadIdx.x & 31;
  const int gw = blockIdx.x * 4 + wave;            // 0..575
  const int b = gw / 9, t = gw % 9;
  const int tm = t / 3, tn = t % 3;                // 3 x 3 tiles per sample
  const int half = lane >> 4, tl = lane & 15;
  const float* mi = MM + ((size_t)b * KTOT + tm * 16 + tl) * MDIM + 2 * half;
  const float* mj = MM + ((size_t)b * KTOT + tn * 16 + tl) * MDIM + 2 * half;
  v8f acc = {};
  for (int k = 0; k < MDIM; k += 4) {
    v2f a = *(const v2f*)(mi + k);
    v2f bv = *(const v2f*)(mj + k);
    acc = wmma4(a, bv, acc);
  }
#pragma unroll
  for (int i = 0; i < 8; ++i) {
    int row = tm * 16 + i + 8 * half;
    int col = tn * 16 + tl;
    float v = acc[i];
    if (row == col) v += RIDGE;
    G[((size_t)b * KTOT + row) * KTOT + col] = v;
  }
}

// ---------------------------------------------------------------- rhs_b = MM_b @ y_b
__global__ void k_rhs(const float* __restrict__ MM, const float* __restrict__ Y,
                      float* __restrict__ RHS) {
  const int idx = blockIdx.x * blockDim.x + threadIdx.x;
  if (idx >= BATCH * KTOT) return;
  const int b = idx / KTOT, i = idx % KTOT;
  const float* m = MM + ((size_t)b * KTOT + i) * MDIM;
  const float* y = Y + (size_t)b * MDIM;
  float s = 0.0f;
  for (int k = 0; k < MDIM; ++k) s += m[k] * y[k];
  RHS[idx] = s;
}

// ---------------------------------------------------------------- Cholesky solve  G z = rhs
__global__ void k_chol(const float* __restrict__ G, const float* __restrict__ RHS,
                       float* __restrict__ Z) {
  __shared__ float g[KTOT * KTOT];
  __shared__ float r[KTOT];
  const int b = blockIdx.x;
  for (int idx = threadIdx.x; idx < KTOT * KTOT; idx += blockDim.x)
    g[idx] = G[(size_t)b * KTOT * KTOT + idx];
  if (threadIdx.x < KTOT) r[threadIdx.x] = RHS[b * KTOT + threadIdx.x];
  __syncthreads();
  for (int kk = 0; kk < KTOT; ++kk) {
    if (threadIdx.x == 0) g[kk * KTOT + kk] = sqrtf(fmaxf(g[kk * KTOT + kk], 1e-20f));
    __syncthreads();
    const float dk = g[kk * KTOT + kk];
    for (int i = kk + 1 + threadIdx.x; i < KTOT; i += blockDim.x) g[i * KTOT + kk] /= dk;
    __syncthreads();
    const int rem = KTOT - kk - 1;
    for (int idx = threadIdx.x; idx < rem * rem; idx += blockDim.x) {
      int i = kk + 1 + idx / rem;
      int j = kk + 1 + idx % rem;
      if (j <= i) g[i * KTOT + j] -= g[i * KTOT + kk] * g[j * KTOT + kk];
    }
    __syncthreads();
  }
  if (threadIdx.x == 0) {
    for (int i = 0; i < KTOT; ++i) {               // forward  L y = rhs
      float s = r[i];
      for (int j = 0; j < i; ++j) s -= g[i * KTOT + j] * r[j];
      r[i] = s / g[i * KTOT + i];
    }
    for (int i = KTOT - 1; i >= 0; --i) {          // backward L^T z = y
      float s = r[i];
      for (int j = i + 1; j < KTOT; ++j) s -= g[j * KTOT + i] * r[j];
      r[i] = s / g[i * KTOT + i];
    }
  }
  __syncthreads();
  if (threadIdx.x < KTOT) Z[b * KTOT + threadIdx.x] = r[threadIdx.x];
}

// ---------------------------------------------------------------- u = P^T z ; SC = |W*u|
__global__ void k_u(const float* __restrict__ P, const float* __restrict__ Z,
                    const float* __restrict__ W, float* __restrict__ U,
                    float* __restrict__ SC) {
  const int b = blockIdx.y;
  const int n = blockIdx.x * 256 + threadIdx.x;
  __shared__ float z[KTOT];
  if (threadIdx.x < KTOT) z[threadIdx.x] = Z[b * KTOT + threadIdx.x];
  __syncthreads();
  const float* pb = P + (size_t)b * KTOT * NDIM + n;
  float acc = 0.0f;
#pragma unroll 8
  for (int i = 0; i < KTOT; ++i) acc += pb[(size_t)i * NDIM] * z[i];
  U[(size_t)b * NDIM + n] = acc;
  SC[(size_t)b * NDIM + n] = fabsf(W[n] * acc);
}

// ---------------------------------------------------------------- X = colsum(Q) * u   (Q = P rows 0..15)
__global__ void k_hx(const float* __restrict__ P, const float* __restrict__ U,
                     float* __restrict__ X) {
  const int b = blockIdx.y;
  const int n = blockIdx.x * 256 + threadIdx.x;
  const float* pb = P + (size_t)b * KTOT * NDIM + n;
  float q = 0.0f;
#pragma unroll
  for (int i = 0; i < SSP; ++i) q += pb[(size_t)i * NDIM];
  X[(size_t)b * NDIM + n] = q * U[(size_t)b * NDIM + n];
}

// ----------------------------------------------------------------
extern "C" void kernel_launch(void* const* d_in, const int* in_sizes, int n_in,
                              void* d_out, int out_size, void* d_ws, size_t ws_size,
                              hipStream_t stream) {
  (void)in_sizes; (void)n_in; (void)out_size; (void)ws_size;
  const float* Y = (const float*)d_in[0];   // (64, 512)
  const float* A = (const float*)d_in[1];   // (512, 2048)
  const float* W = (const float*)d_in[2];   // (2048,)
  float* X = (float*)d_out;                 // (64, 2048)

  float* ws = (float*)d_ws;
  float* R   = ws; ws += (size_t)BATCH * MDIM;          // residuals
  float* SC  = ws; ws += (size_t)BATCH * NDIM;          // scores
  float* CC  = ws; ws += (size_t)BATCH * NDIM;          // pairwise C
  float* P   = ws; ws += (size_t)BATCH * KTOT * NDIM;   // [Q_prev(16); P_tilde(32)]
  float* MM  = ws; ws += (size_t)BATCH * KTOT * MDIM;   // M_b = P_b A^T
  float* G   = ws; ws += (size_t)BATCH * KTOT * KTOT;   // Gram
  float* RHS = ws; ws += (size_t)BATCH * KTOT;
  float* Z   = ws; ws += (size_t)BATCH * KTOT;
  float* U   = ws; ws += (size_t)BATCH * NDIM;

  k_init<<<256, 256, 0, stream>>>(X, P);
  for (int layer = 0; layer < 2; ++layer) {
    k_residual<<<8, 128, 0, stream>>>(X, A, Y, R);
    k_scores<<<32, 128, 0, stream>>>(R, A, W, SC);
    k_pairwise<<<BATCH, 256, 0, stream>>>(SC, CC);
    k_softsort<<<BATCH * 2 * SSP, 256, 0, stream>>>(SC, CC, P, 2 * SSP, SSP);
    k_pmat<<<384, 128, 0, stream>>>(P, A, MM);
    k_gram<<<144, 128, 0, stream>>>(MM, G);
    k_rhs<<<12, 256, 0, stream>>>(MM, Y, RHS);
    k_chol<<<BATCH, 64, 0, stream>>>(G, RHS, Z);
    k_u<<<dim3(NDIM / 256, BATCH), 256, 0, stream>>>(P, Z, W, U, SC);
    k_pairwise<<<BATCH, 256, 0, stream>>>(SC, CC);
    k_softsort<<<BATCH * SSP, 256, 0, stream>>>(SC, CC, P, SSP, 0);
    k_hx<<<dim3(NDIM / 256, BATCH), 256, 0, stream>>>(P, U, X);
  }
}